// IRNetLSTMEncoder_1116691497403
// MI455X (gfx1250) — compile-verified
//
#include <hip/hip_runtime.h>

// ---------------------------------------------------------------------------
// IRNet LSTM encoder for MI455X (gfx1250), fp32 path via V_WMMA_F32_16X16X4_F32
// ---------------------------------------------------------------------------

typedef __attribute__((ext_vector_type(2))) float v2f;
typedef __attribute__((ext_vector_type(8))) float v8f;

#define B_SZ   256
#define S_LEN  128
#define D_EMB  300
#define H_SZ   150
#define KP     452      // 300 (x) + 152 (h padded), multiple of 4
#define NP     608      // 600 padded to 38*16
#define NTILES (NP/16)  // 38
#define LDA    454      // LDS stride for A tile (even -> float2 aligned)
#define LDG    608      // LDS stride for gate buffer

__device__ inline v8f wmma_f32_4(v2f a, v2f b, v8f c) {
  // D = A(16x4, f32) * B(4x16, f32) + C(16x16, f32)
  return __builtin_amdgcn_wmma_f32_16x16x4_f32(
      /*neg_a=*/false, a, /*neg_b=*/false, b,
      /*c_mod=*/(short)0, c, /*reuse_a=*/false, /*reuse_b=*/false);
}

__device__ inline float sigf(float x) { return 1.0f / (1.0f + __expf(-x)); }

// ---------------------------------------------------------------------------
// Pack [Wx; Wh] into K-pair-interleaved layout:
//   wp[dir][ ((k>>1)*NP + n)*2 + (k&1) ] = W[k][n]
// rows 0..299 = Wx, 300..449 = Wh, 450..451 = 0; cols >= 600 are 0.
// ---------------------------------------------------------------------------
__global__ void pack_kernel(const float* __restrict__ Wx_f,
                            const float* __restrict__ Wh_f,
                            const float* __restrict__ Wx_b,
                            const float* __restrict__ Wh_b,
                            float* __restrict__ wp) {
  int idx = blockIdx.x * 256 + threadIdx.x;
  const int per_dir = KP * NP;
  if (idx >= 2 * per_dir) return;
  int dir = idx / per_dir;
  int r   = idx % per_dir;
  int k   = r / NP;
  int n   = r % NP;
  const float* Wx = dir ? Wx_b : Wx_f;
  const float* Wh = dir ? Wh_b : Wh_f;
  float v = 0.0f;
  if (n < 600) {
    if (k < 300)      v = Wx[k * 600 + n];
    else if (k < 450) v = Wh[(k - 300) * 600 + n];
  }
  wp[(size_t)dir * per_dir + ((size_t)(k >> 1) * NP + n) * 2 + (k & 1)] = v;
}

// ---------------------------------------------------------------------------
// Row L2 norms clamped to eps (rows of length 300).
// ---------------------------------------------------------------------------
__global__ void norm_kernel(const float* __restrict__ x,
                            float* __restrict__ out, int rows) {
  int i = blockIdx.x * 256 + threadIdx.x;
  if (i >= rows) return;
  const float* p = x + (size_t)i * D_EMB;
  float s = 0.0f;
  for (int d = 0; d < D_EMB; ++d) { float v = p[d]; s += v * v; }
  out[i] = fmaxf(sqrtf(s), 1e-8f);
}

// ---------------------------------------------------------------------------
// BiLSTM scan. Grid = 32 blocks: blockIdx/16 = direction, blockIdx%16 = batch
// tile of 16 rows. 256 threads = 8 waves; each wave owns ~5 of the 38 N-tiles.
// A tile = [x_t (300) | h (150, pad 152)] staged in LDS; B from packed weights.
// ---------------------------------------------------------------------------
__global__ void __launch_bounds__(256)
lstm_scan_kernel(const float* __restrict__ src_emb,   // [B,S,300]
                 const float* __restrict__ wpack,     // [2][KP*NP] packed
                 const float* __restrict__ bias_f,    // [600]
                 const float* __restrict__ bias_b,    // [600]
                 float* __restrict__ out_src,         // [B,S,300]
                 float* __restrict__ out_cell)        // [B,300]
{
  __shared__ float As[16 * LDA];      // A tile: x | h
  __shared__ float gbuf[16 * LDG];    // gate pre-activations
  __shared__ float cbuf[16 * 152];    // cell state

  int dir = blockIdx.x >> 4;
  int m0  = (blockIdx.x & 15) * 16;
  const float* bias = dir ? bias_b : bias_f;
  const v2f* Wp2 = (const v2f*)(wpack + (size_t)dir * (KP * NP));

  int tid  = threadIdx.x;
  int lane = tid & 31;
  int wv   = tid >> 5;          // wave id 0..7
  int hi   = (lane >> 4) & 1;   // high half of wave -> K offset +2
  int ln   = lane & 15;         // matrix row (A) / col (B,D)

  // zero h region (cols 300..451) and cell state
  for (int idx = tid; idx < 16 * 152; idx += 256) {
    int m = idx / 152, j = idx % 152;
    As[m * LDA + 300 + j] = 0.0f;
    cbuf[m * 152 + j] = 0.0f;
  }
  __syncthreads();

  for (int step = 0; step < S_LEN; ++step) {
    int t = dir ? (S_LEN - 1 - step) : step;

    // stage x_t tile into A[:, 0:300]
    for (int idx = tid; idx < 16 * D_EMB; idx += 256) {
      int m = idx / D_EMB, d = idx % D_EMB;
      As[m * LDA + d] = src_emb[((size_t)(m0 + m) * S_LEN + t) * D_EMB + d];
    }
    __syncthreads();

    // GEMM: [16 x 452] * [452 x 608] via f32 WMMA
    for (int nt = wv; nt < NTILES; nt += 8) {
      int n = nt * 16 + ln;
      v8f acc = {};
      for (int k0 = 0; k0 < KP; k0 += 4) {
        v2f a = *(const v2f*)&As[ln * LDA + k0 + hi * 2];
        v2f b = Wp2[(size_t)((k0 >> 1) + hi) * NP + n];
        acc = wmma_f32_4(a, b, acc);
      }
      float bn = (n < 600) ? bias[n] : 0.0f;
#pragma unroll
      for (int r = 0; r < 8; ++r)
        gbuf[(r + hi * 8) * LDG + n] = acc[r] + bn;
    }
    __syncthreads();

    // gate nonlinearity, state update, write h
    for (int idx = tid; idx < 16 * H_SZ; idx += 256) {
      int m = idx / H_SZ, j = idx % H_SZ;
      float gi = gbuf[m * LDG + j];
      float gf = gbuf[m * LDG + 150 + j];
      float gg = gbuf[m * LDG + 300 + j];
      float go = gbuf[m * LDG + 450 + j];
      float c  = sigf(gf) * cbuf[m * 152 + j] + sigf(gi) * tanhf(gg);
      float h  = sigf(go) * tanhf(c);
      cbuf[m * 152 + j] = c;
      As[m * LDA + 300 + j] = h;
      out_src[((size_t)(m0 + m) * S_LEN + t) * D_EMB + dir * H_SZ + j] = h;
    }
    __syncthreads();
  }

  // final cell state -> last_cell
  for (int idx = tid; idx < 16 * H_SZ; idx += 256) {
    int m = idx / H_SZ, j = idx % H_SZ;
    out_cell[(size_t)(m0 + m) * D_EMB + dir * H_SZ + j] = cbuf[m * 152 + j];
  }
}

// ---------------------------------------------------------------------------
// sim[b][c][s] = (cmp[b][c] . src[b][s]) / (cn[b][c] * sn[b][s])
// One WG (4 waves) per 16-row c-tile; K = 300 (75 WMMA k-steps), N = 128.
// ---------------------------------------------------------------------------
__global__ void __launch_bounds__(128)
sim_kernel(const float* __restrict__ cmp,   // [B, C, 300]
           const float* __restrict__ src,   // [B, 128, 300]
           const float* __restrict__ cn,    // [B*C]
           const float* __restrict__ sn,    // [B*128]
           float* __restrict__ sim,         // [B*C*128]
           int C)
{
  __shared__ float Asm[16 * 302];
  int nct = C >> 4;
  int bb  = blockIdx.x / nct;
  int c0  = (blockIdx.x % nct) * 16;
  int tid = threadIdx.x, lane = tid & 31, wv = tid >> 5;
  int hi  = (lane >> 4) & 1, ln = lane & 15;

  const float* cptr = cmp + ((size_t)bb * C + c0) * D_EMB;
  const float* sptr = src + (size_t)bb * S_LEN * D_EMB;

  for (int idx = tid; idx < 16 * D_EMB; idx += 128) {
    int m = idx / D_EMB, d = idx % D_EMB;
    Asm[m * 302 + d] = cptr[(size_t)m * D_EMB + d];
  }
  __syncthreads();

  for (int nt = wv; nt < 8; nt += 4) {
    int s = nt * 16 + ln;
    v8f acc = {};
    for (int k0 = 0; k0 < D_EMB; k0 += 4) {
      v2f a = *(const v2f*)&Asm[ln * 302 + k0 + hi * 2];
      v2f b = *(const v2f*)&sptr[(size_t)s * D_EMB + k0 + hi * 2];
      acc = wmma_f32_4(a, b, acc);
    }
    float sns = sn[bb * S_LEN + s];
#pragma unroll
    for (int r = 0; r < 8; ++r) {
      int m = r + hi * 8;
      float cnv = cn[(size_t)bb * C + c0 + m];
      sim[((size_t)bb * C + c0 + m) * S_LEN + s] = acc[r] / (cnv * sns);
    }
  }
}

// ---------------------------------------------------------------------------
// out[b][c][n] = emb + sim[b][c][:] . src_enc[b][:][n] + (hot @ W_hot + b_hot)
// K = 128 (32 WMMA k-steps), N = 300 (19 tiles of 16, last partially masked).
// ---------------------------------------------------------------------------
__global__ void __launch_bounds__(128)
ctx_kernel(const float* __restrict__ sim,   // [B*C*128]
           const float* __restrict__ senc,  // [B*128*300] (src_encoding)
           const float* __restrict__ emb,   // [B*C*300]
           const float* __restrict__ hot,   // [B*C*hk]
           const float* __restrict__ Whot,  // [hk*300]
           const float* __restrict__ bhot,  // [300]
           float* __restrict__ out,         // [B*C*300]
           int C, int hk)
{
  __shared__ float Asm[16 * 130];
  __shared__ float hbuf[16 * 9];
  int nct = C >> 4;
  int bb  = blockIdx.x / nct;
  int c0  = (blockIdx.x % nct) * 16;
  int tid = threadIdx.x, lane = tid & 31, wv = tid >> 5;
  int hi  = (lane >> 4) & 1, ln = lane & 15;

  for (int idx = tid; idx < 16 * S_LEN; idx += 128) {
    int m = idx >> 7, s = idx & 127;
    Asm[m * 130 + s] = sim[((size_t)bb * C + c0 + m) * S_LEN + s];
  }
  for (int idx = tid; idx < 16 * hk; idx += 128) {
    int m = idx / hk, k = idx % hk;
    hbuf[m * 9 + k] = hot[((size_t)bb * C + c0 + m) * hk + k];
  }
  __syncthreads();

  const float* sb = senc + (size_t)bb * S_LEN * D_EMB;
  for (int nt = wv; nt < 19; nt += 4) {
    int n = nt * 16 + ln;
    v8f acc = {};
    for (int k0 = 0; k0 < S_LEN; k0 += 4) {
      v2f a = *(const v2f*)&Asm[ln * 130 + k0 + hi * 2];
      v2f b;
      if (n < D_EMB) {
        b.x = sb[(size_t)(k0 + hi * 2) * D_EMB + n];
        b.y = sb[(size_t)(k0 + hi * 2 + 1) * D_EMB + n];
      } else {
        b.x = 0.0f; b.y = 0.0f;
      }
      acc = wmma_f32_4(a, b, acc);
    }
    if (n < D_EMB) {
      float wc[9];
      for (int k = 0; k < hk; ++k) wc[k] = Whot[k * D_EMB + n];
      float bn = bhot[n];
#pragma unroll
      for (int r = 0; r < 8; ++r) {
        int m = r + hi * 8;
        float hsum = 0.0f;
        for (int k = 0; k < hk; ++k) hsum += hbuf[m * 9 + k] * wc[k];
        size_t off = ((size_t)bb * C + c0 + m) * D_EMB + n;
        out[off] = emb[off] + acc[r] + bn + hsum;
      }
    }
  }
}

// ---------------------------------------------------------------------------
extern "C" void kernel_launch(void* const* d_in, const int* in_sizes, int n_in,
                              void* d_out, int out_size, void* d_ws, size_t ws_size,
                              hipStream_t stream) {
  const float* src_emb = (const float*)d_in[0];
  const float* col_emb = (const float*)d_in[1];
  const float* tab_emb = (const float*)d_in[2];
  const float* col_hot = (const float*)d_in[3];
  const float* tab_hot = (const float*)d_in[4];
  const float* Wx_f = (const float*)d_in[5];
  const float* Wh_f = (const float*)d_in[6];
  const float* b_f  = (const float*)d_in[7];
  const float* Wx_b = (const float*)d_in[8];
  const float* Wh_b = (const float*)d_in[9];
  const float* b_b  = (const float*)d_in[10];
  const float* W_col = (const float*)d_in[11];
  const float* b_col = (const float*)d_in[12];
  const float* W_tab = (const float*)d_in[13];
  const float* b_tab = (const float*)d_in[14];

  float* out = (float*)d_out;
  float* out_src  = out;                          // [256,128,300]
  float* out_col  = out + 9830400;                // [256,256,300]
  float* out_tab  = out + 29491200;               // [256,64,300]
  float* out_cell = out + 34406400;               // [256,300]

  float* ws      = (float*)d_ws;
  float* wpack   = ws;                            // 2*452*608 = 549,632
  float* sn      = wpack + 2 * KP * NP;           // 32,768
  float* cn_col  = sn + B_SZ * S_LEN;             // 65,536
  float* cn_tab  = cn_col + B_SZ * 256;           // 16,384
  float* sim_col = cn_tab + B_SZ * 64;            // 8,388,608
  float* sim_tab = sim_col + (size_t)B_SZ * 256 * S_LEN; // 2,097,152

  // 1) pack weights
  pack_kernel<<<(2 * KP * NP + 255) / 256, 256, 0, stream>>>(
      Wx_f, Wh_f, Wx_b, Wh_b, wpack);

  // 2) norms
  norm_kernel<<<(B_SZ * S_LEN + 255) / 256, 256, 0, stream>>>(src_emb, sn, B_SZ * S_LEN);
  norm_kernel<<<(B_SZ * 256 + 255) / 256, 256, 0, stream>>>(col_emb, cn_col, B_SZ * 256);
  norm_kernel<<<(B_SZ * 64 + 255) / 256, 256, 0, stream>>>(tab_emb, cn_tab, B_SZ * 64);

  // 3) BiLSTM scan (writes src_encoding + last_cell)
  lstm_scan_kernel<<<32, 256, 0, stream>>>(src_emb, wpack, b_f, b_b, out_src, out_cell);

  // 4) cosine similarities
  sim_kernel<<<B_SZ * (256 / 16), 128, 0, stream>>>(col_emb, src_emb, cn_col, sn, sim_col, 256);
  sim_kernel<<<B_SZ * (64 / 16), 128, 0, stream>>>(tab_emb, src_emb, cn_tab, sn, sim_tab, 64);

  // 5) context GEMM + type-encoder epilogue
  ctx_kernel<<<B_SZ * (256 / 16), 128, 0, stream>>>(
      sim_col, out_src, col_emb, col_hot, W_col, b_col, out_col, 256, 9);
  ctx_kernel<<<B_SZ * (64 / 16), 128, 0, stream>>>(
      sim_tab, out_src, tab_emb, tab_hot, W_tab, b_tab, out_tab, 64, 5);
}